// LightGCN_31636729103187
// MI455X (gfx1250) — compile-verified
//
#include <hip/hip_runtime.h>
#include <hip/hip_bf16.h>
#include <stdint.h>

#define NUM_USERS 100000
#define NUM_ITEMS 50000
#define N_NODES   150000
#define EMBED_DIM 64
#define NELEM     (N_NODES * EMBED_DIM)        // 9,600,000 floats
#define UELEM     (NUM_USERS * EMBED_DIM)      // 6,400,000 floats (mult of 4)

// ---------------------------------------------------------------------------
// init: h_a = concat(user_emb, item_emb); out = h_a; h_b = 0
// ---------------------------------------------------------------------------
__global__ __launch_bounds__(256) void lgcn_init(const float* __restrict__ user_emb,
                                                 const float* __restrict__ item_emb,
                                                 float* __restrict__ h_a,
                                                 float* __restrict__ h_b,
                                                 float* __restrict__ out) {
    int base = (blockIdx.x * 256 + threadIdx.x) * 4;
    if (base >= NELEM) return;
    float4 v;
    if (base < UELEM) v = *(const float4*)(user_emb + base);
    else              v = *(const float4*)(item_emb + (base - UELEM));
    *(float4*)(h_a + base) = v;
    *(float4*)(out + base) = v;
    *(float4*)(h_b + base) = make_float4(0.f, 0.f, 0.f, 0.f);
}

// ---------------------------------------------------------------------------
// accum_zero: sum += src; zbuf = 0   (zbuf is the next layer's output buffer)
// ---------------------------------------------------------------------------
__global__ __launch_bounds__(256) void lgcn_accum_zero(float* __restrict__ sum,
                                                       const float* __restrict__ src,
                                                       float* __restrict__ zbuf) {
    int base = (blockIdx.x * 256 + threadIdx.x) * 4;
    if (base >= NELEM) return;
    float4 s = *(float4*)(sum + base);
    float4 a = *(const float4*)(src + base);
    s.x += a.x; s.y += a.y; s.z += a.z; s.w += a.w;
    *(float4*)(sum + base) = s;
    *(float4*)(zbuf + base) = make_float4(0.f, 0.f, 0.f, 0.f);
}

// ---------------------------------------------------------------------------
// final: sum = (sum + src) * 0.25f
// ---------------------------------------------------------------------------
__global__ __launch_bounds__(256) void lgcn_final(float* __restrict__ sum,
                                                  const float* __restrict__ src) {
    int base = (blockIdx.x * 256 + threadIdx.x) * 4;
    if (base >= NELEM) return;
    float4 s = *(float4*)(sum + base);
    float4 a = *(const float4*)(src + base);
    s.x = (s.x + a.x) * 0.25f; s.y = (s.y + a.y) * 0.25f;
    s.z = (s.z + a.z) * 0.25f; s.w = (s.w + a.w) * 0.25f;
    *(float4*)(sum + base) = s;
}

// ---------------------------------------------------------------------------
// SpMM: y[r,:] += v * x[c,:]   one wave32 per nnz entry; lane l covers
// dims l and l+32 (two coalesced 128B transactions per nnz).
// COO chunk (256 entries) staged into LDS with gfx1250 async-to-LDS copies.
// ---------------------------------------------------------------------------
__global__ __launch_bounds__(256, 4) void lgcn_spmm(const float* __restrict__ vals,
                                                    const int*   __restrict__ rows,
                                                    const int*   __restrict__ cols,
                                                    const float* __restrict__ x,
                                                    float*       __restrict__ y,
                                                    int nnz) {
    __shared__ int sRow[256];
    __shared__ int sCol[256];
    __shared__ int sVal[256];

    const int tid  = threadIdx.x;
    const int g    = blockIdx.x * 256 + tid;

    // --- stage this block's 256 COO entries into LDS via async DMA path ---
    if (g < nnz) {
        unsigned ldsR = (unsigned)(uintptr_t)&sRow[tid];
        unsigned ldsC = (unsigned)(uintptr_t)&sCol[tid];
        unsigned ldsV = (unsigned)(uintptr_t)&sVal[tid];
        unsigned off  = (unsigned)g * 4u;
        asm volatile("global_load_async_to_lds_b32 %0, %1, %2"
                     :: "v"(ldsR), "v"(off), "s"(rows) : "memory");
        asm volatile("global_load_async_to_lds_b32 %0, %1, %2"
                     :: "v"(ldsC), "v"(off), "s"(cols) : "memory");
        asm volatile("global_load_async_to_lds_b32 %0, %1, %2"
                     :: "v"(ldsV), "v"(off), "s"(vals) : "memory");
    }
    asm volatile("s_wait_asynccnt 0" ::: "memory");

    const int lane  = tid & 31;
    const int wbase = tid & ~31;          // this wave's 32 staged entries
    const int blk   = blockIdx.x * 256;

    #pragma unroll 4
    for (int j = 0; j < 32; ++j) {
        const int e = wbase + j;
        if (blk + e >= nnz) break;
        // uniform per wave: broadcast read from LDS
        const int   r = sRow[e];
        const int   c = sCol[e];
        const float v = __int_as_float(sVal[e]);

        const float x0 = x[c * EMBED_DIM + lane];
        const float x1 = x[c * EMBED_DIM + 32 + lane];
        // native global_atomic_add_f32 (L2-resident output, 38.4MB < 192MB L2)
        unsafeAtomicAdd(&y[r * EMBED_DIM + lane],      v * x0);
        unsafeAtomicAdd(&y[r * EMBED_DIM + 32 + lane], v * x1);
    }
}

// ---------------------------------------------------------------------------
extern "C" void kernel_launch(void* const* d_in, const int* in_sizes, int n_in,
                              void* d_out, int out_size, void* d_ws, size_t ws_size,
                              hipStream_t stream) {
    const float* user_emb = (const float*)d_in[0];
    const float* item_emb = (const float*)d_in[1];
    const float* vals     = (const float*)d_in[2];
    const int*   rows     = (const int*)d_in[3];
    const int*   cols     = (const int*)d_in[4];
    float* out = (float*)d_out;

    float* h_a = (float*)d_ws;            // 38.4 MB
    float* h_b = h_a + NELEM;             // 38.4 MB
    const int nnz = in_sizes[2];

    const dim3 blk(256);
    const dim3 egrid((NELEM / 4 + 255) / 256);      // elementwise kernels (float4)
    const dim3 sgrid((nnz + 255) / 256);            // one block per 256 nnz

    // layer 0: h_a = all_emb, out = all_emb, h_b zeroed
    lgcn_init<<<egrid, blk, 0, stream>>>(user_emb, item_emb, h_a, h_b, out);

    // layer 1: h_b = A @ h_a ; out += h_b ; zero h_a for next layer
    lgcn_spmm<<<sgrid, blk, 0, stream>>>(vals, rows, cols, h_a, h_b, nnz);
    lgcn_accum_zero<<<egrid, blk, 0, stream>>>(out, h_b, h_a);

    // layer 2: h_a = A @ h_b ; out += h_a ; zero h_b
    lgcn_spmm<<<sgrid, blk, 0, stream>>>(vals, rows, cols, h_b, h_a, nnz);
    lgcn_accum_zero<<<egrid, blk, 0, stream>>>(out, h_a, h_b);

    // layer 3: h_b = A @ h_a ; out = (out + h_b) / 4
    lgcn_spmm<<<sgrid, blk, 0, stream>>>(vals, rows, cols, h_a, h_b, nnz);
    lgcn_final<<<egrid, blk, 0, stream>>>(out, h_b);
}